// EMDLoss_85237920956692
// MI455X (gfx1250) — compile-verified
//
#include <hip/hip_runtime.h>

// ---------------------------------------------------------------------------
// Sinkhorn EMD on gfx1250 (MI455X).
//   x,y : [16][3][1024] f32.  One workgroup (32 wave32) per batch.
//   Kernel matrix M_ij = g_ij + a_i + b_j with g = (2/eps) X.Y^T computed per
//   16x16 tile by V_WMMA_F32_16X16X4_F32.  D=3 is padded to K=4 and the spare
//   K slot carries {1.0 on the A side, per-column coefficient on the B side},
//   so one WMMA with an inline-0 accumulator yields g_ij + coef_j directly
//   (no accumulator broadcast movs, no coef LDS load).
//   Row-logsumexp is two-pass (max sweep, then exp sweep) to halve the
//   v_exp_f32 pressure on the TRANS pipe -- the throughput limiter, since the
//   whole solve is LDS-resident.  Grid-wide batch-coupled early stop via a
//   monotonic atomic counter barrier in workspace.
// ---------------------------------------------------------------------------

#define NPTS      1024
#define NTILES    64          // 1024 / 16
#define BATCH     16
#define MAX_IT    200
#define EPS       1e-3f
#define INV_EPS   1000.0f
#define AB_SCALE  2000.0f     // 2/eps folded into x side
#define THRESH    1e-1f

typedef __attribute__((ext_vector_type(2))) float v2f;
typedef __attribute__((ext_vector_type(8))) float v8f;

struct Ctrl {
  unsigned int count;          // monotonic grid-barrier counter
  unsigned int pad[31];
  float gerr[MAX_IT];          // per-iteration summed |du| across batches
  float cost[BATCH];           // per-batch transport cost
};

__device__ __forceinline__ v8f wmma_k4z(v2f a, v2f b) {
  // D = A(16x4,f32) * B(4x16,f32) + 0  (inline-zero accumulator)
  v8f z = {};
  return __builtin_amdgcn_wmma_f32_16x16x4_f32(
      /*neg_a=*/false, a, /*neg_b=*/false, b,
      /*c_mod=*/(short)0, z, /*reuse_a=*/false, /*reuse_b=*/false);
}

// One Sinkhorn half-update:
//   wvec[row] = logc + nrm[row] - eps * LSE_col( SA[row,0:3].SB[col,0:3] + SB[col,3] )
// SA slot3 must hold 1.0, SB slot3 holds the per-column coefficient.
__device__ __forceinline__ void sinkhorn_half(
    const float* SA, const float* SB,
    const float* nrm, float* wvec, float logc, float* errLds,
    int lane, int wid)
{
  const int lm   = lane & 15;
  const int half = lane >> 4;
  const v2f* SA2 = (const v2f*)SA;   // 2 x v2f per padded point
  const v2f* SB2 = (const v2f*)SB;

  for (int strip = 0; strip < 2; ++strip) {
    const int ti = wid + strip * 32;                 // tile-row 0..63
    const v2f a  = SA2[(ti * 16 + lm) * 2 + half];   // A: lane -> row lm, K-half

    // ---- pass 1: row maxima (VALU only, no exp) ----
    float m[8];
#pragma unroll
    for (int r = 0; r < 8; ++r) m[r] = -__builtin_inff();
    for (int tj = 0; tj < NTILES; ++tj) {
      const v2f bv = SB2[(tj * 16 + lm) * 2 + half];
      v8f d = wmma_k4z(a, bv);                       // d = g_tile + coef_j
#pragma unroll
      for (int r = 0; r < 8; ++r) m[r] = fmaxf(m[r], d[r]);
    }
#pragma unroll
    for (int r = 0; r < 8; ++r) {                    // merge across 16-lane half
      float mr = m[r];
#pragma unroll
      for (int off = 8; off >= 1; off >>= 1)
        mr = fmaxf(mr, __shfl_xor(mr, off, 32));
      m[r] = mr;                                     // exact row max, all lanes
    }

    // ---- pass 2: sum exp(d - rowmax) (1 exp per element) ----
    float s[8];
#pragma unroll
    for (int r = 0; r < 8; ++r) s[r] = 0.0f;
    for (int tj = 0; tj < NTILES; ++tj) {
      const v2f bv = SB2[(tj * 16 + lm) * 2 + half];
      v8f d = wmma_k4z(a, bv);
#pragma unroll
      for (int r = 0; r < 8; ++r) s[r] += __expf(d[r] - m[r]);
    }
#pragma unroll
    for (int r = 0; r < 8; ++r) {                    // plain-add lane merge
      float sr = s[r];
#pragma unroll
      for (int off = 8; off >= 1; off >>= 1)
        sr += __shfl_xor(sr, off, 32);
      s[r] = sr;
    }

    if (lm == 0) {                                   // lanes 0 and 16 write
      float e = 0.0f;
#pragma unroll
      for (int r = 0; r < 8; ++r) {
        const int   row = ti * 16 + half * 8 + r;
        const float L   = m[r] + __logf(s[r]);
        const float nw  = logc + nrm[row] - EPS * L;
        e += fabsf(nw - wvec[row]);
        wvec[row] = nw;
      }
      if (errLds) atomicAdd(errLds, e);              // ds_add_f32
    }
  }
}

__global__ void __launch_bounds__(1024)
sinkhorn_kernel(const float* __restrict__ X, const float* __restrict__ Y,
                Ctrl* __restrict__ ctrl)
{
  __shared__ float xs[NPTS * 4];   // [x*(2/eps), slot3 = 1.0 or a_i]
  __shared__ float ys[NPTS * 4];   // [y,         slot3 = 1.0 or b_j]
  __shared__ float xn[NPTS], yn[NPTS];
  __shared__ float uu[NPTS], vvv[NPTS];
  __shared__ float wErr;
  __shared__ int   sDone;
  __shared__ float wsum[32];

  const int tid  = threadIdx.x;
  const int lane = tid & 31;
  const int wid  = tid >> 5;
  const int b    = blockIdx.x;

  { // load points, fold scale, compute squared norms
    const int i = tid;
    const float x0 = X[b * 3 * NPTS + 0 * NPTS + i];
    const float x1 = X[b * 3 * NPTS + 1 * NPTS + i];
    const float x2 = X[b * 3 * NPTS + 2 * NPTS + i];
    xs[i * 4 + 0] = x0 * AB_SCALE; xs[i * 4 + 1] = x1 * AB_SCALE;
    xs[i * 4 + 2] = x2 * AB_SCALE; xs[i * 4 + 3] = 1.0f;
    xn[i] = x0 * x0 + x1 * x1 + x2 * x2;
    const float y0 = Y[b * 3 * NPTS + 0 * NPTS + i];
    const float y1 = Y[b * 3 * NPTS + 1 * NPTS + i];
    const float y2 = Y[b * 3 * NPTS + 2 * NPTS + i];
    ys[i * 4 + 0] = y0; ys[i * 4 + 1] = y1; ys[i * 4 + 2] = y2;
    ys[i * 4 + 3] = 1.0f;
    yn[i] = y0 * y0 + y1 * y1 + y2 * y2;
    uu[i] = 0.0f; vvv[i] = 0.0f;
  }
  __syncthreads();

  const float logc = EPS * __logf(1.0f / 1024.0f + 1e-8f); // eps*log(mu)

  for (int it = 0; it < MAX_IT; ++it) {
    xs[tid * 4 + 3] = 1.0f;                                // A-side pad
    ys[tid * 4 + 3] = (vvv[tid] - yn[tid]) * INV_EPS;      // b_j into B slot3
    if (tid == 0) wErr = 0.0f;
    __syncthreads();
    sinkhorn_half(xs, ys, xn, uu, logc, &wErr, lane, wid); // u update
    __syncthreads();
    xs[tid * 4 + 3] = (uu[tid] - xn[tid]) * INV_EPS;       // a_i into B slot3
    ys[tid * 4 + 3] = 1.0f;                                // A-side pad
    __syncthreads();
    sinkhorn_half(ys, xs, yn, vvv, logc, nullptr, lane, wid); // v update
    __syncthreads();

    if (tid == 0) {                                        // batch-coupled stop
      atomicAdd(&ctrl->gerr[it], wErr);
      __threadfence();
      atomicAdd(&ctrl->count, 1u);
      const unsigned tgt = (unsigned)(BATCH * (it + 1));
      volatile unsigned* vc = &ctrl->count;
      while (*vc < tgt) __builtin_amdgcn_s_sleep(1);
      __threadfence();
      const float tot = *((volatile float*)&ctrl->gerr[it]);
      sDone = (tot * (1.0f / BATCH) < THRESH) ? 1 : 0;
    }
    __syncthreads();
    if (sDone) break;
  }

  // transport cost: sum_ij exp(M_ij) * C_ij, C = xn + yn - eps*g
  xs[tid * 4 + 3] = 1.0f;
  ys[tid * 4 + 3] = (vvv[tid] - yn[tid]) * INV_EPS;        // b_j
  __syncthreads();

  const int lm   = lane & 15;
  const int half = lane >> 4;
  const v2f* xs2 = (const v2f*)xs;
  const v2f* ys2 = (const v2f*)ys;
  float acc = 0.0f;
  for (int strip = 0; strip < 2; ++strip) {
    const int ti = wid + strip * 32;
    const v2f a  = xs2[(ti * 16 + lm) * 2 + half];
    float ar[8], xr[8];
#pragma unroll
    for (int r = 0; r < 8; ++r) {
      const int row = ti * 16 + half * 8 + r;
      xr[r] = xn[row];
      ar[r] = (uu[row] - xr[r]) * INV_EPS;
    }
    for (int tj = 0; tj < NTILES; ++tj) {
      const int   col = tj * 16 + lm;
      const v2f   bv  = ys2[col * 2 + half];
      const float ynj = yn[col];
      const float cv  = ys[col * 4 + 3];                   // b_j
      v8f d = wmma_k4z(a, bv);                             // t = g + b_j
#pragma unroll
      for (int r = 0; r < 8; ++r) {
        const float t  = d[r];
        const float pi = __expf(t + ar[r]);
        const float Cc = xr[r] + ynj - EPS * (t - cv);
        acc += pi * Cc;
      }
    }
  }
#pragma unroll
  for (int off = 16; off >= 1; off >>= 1) acc += __shfl_xor(acc, off, 32);
  if (lane == 0) wsum[wid] = acc;
  __syncthreads();
  if (wid == 0) {
    float t2 = wsum[lane];
#pragma unroll
    for (int off = 16; off >= 1; off >>= 1) t2 += __shfl_xor(t2, off, 32);
    if (lane == 0) ctrl->cost[b] = t2;
  }
}

__global__ void init_kernel(Ctrl* ctrl) {
  const int t = threadIdx.x;
  if (t == 0) ctrl->count = 0u;
  for (int i = t; i < MAX_IT; i += blockDim.x) ctrl->gerr[i] = 0.0f;
  if (t < BATCH) ctrl->cost[t] = 0.0f;
}

__global__ void finalize_kernel(const Ctrl* __restrict__ ctrl, float* __restrict__ out) {
  if (threadIdx.x == 0) {
    float s = 0.0f;
    for (int i = 0; i < BATCH; ++i) s += ctrl->cost[i];
    out[0] = s * (1.0f / BATCH);
  }
}

extern "C" void kernel_launch(void* const* d_in, const int* in_sizes, int n_in,
                              void* d_out, int out_size, void* d_ws, size_t ws_size,
                              hipStream_t stream) {
  (void)in_sizes; (void)n_in; (void)out_size; (void)ws_size;
  const float* x = (const float*)d_in[0];
  const float* y = (const float*)d_in[1];
  Ctrl* ctrl = (Ctrl*)d_ws;
  init_kernel<<<1, 256, 0, stream>>>(ctrl);
  sinkhorn_kernel<<<BATCH, 1024, 0, stream>>>(x, y, ctrl);
  finalize_kernel<<<1, 64, 0, stream>>>(ctrl, (float*)d_out);
}